// RankingSet_50749333569648
// MI455X (gfx1250) — compile-verified
//
#include <hip/hip_runtime.h>
#include <hip/hip_bf16.h>

typedef float v2f __attribute__((ext_vector_type(2)));
typedef float v8f __attribute__((ext_vector_type(8)));

#define D_DIM 512
#define Q_DIM 128
#define MTILE 128          // data rows per block (8 waves x 16 rows)
#define EPS_NORM 1e-12f
#define RTOL_C   1e-5f
#define ATOL_C   1e-8f

// ---------------- kernel 1: normalize queries, per-query thresholds, zero counters --------------
__global__ void rs_prep(const float* __restrict__ queries,
                        const float* __restrict__ truths,
                        float* __restrict__ qn,
                        float* __restrict__ thresh,
                        int* __restrict__ counts) {
    const int q   = blockIdx.x;      // one block per query
    const int tid = threadIdx.x;     // 256 threads
    __shared__ float red[256];

    float sq = 0.f, st = 0.f;
    for (int i = tid; i < D_DIM; i += 256) {
        float a = queries[q * D_DIM + i];
        float b = truths [q * D_DIM + i];
        sq += a * a;
        st += b * b;
    }
    red[tid] = sq; __syncthreads();
    for (int s = 128; s > 0; s >>= 1) { if (tid < s) red[tid] += red[tid + s]; __syncthreads(); }
    const float nq = fmaxf(sqrtf(red[0]), EPS_NORM);
    __syncthreads();

    red[tid] = st; __syncthreads();
    for (int s = 128; s > 0; s >>= 1) { if (tid < s) red[tid] += red[tid + s]; __syncthreads(); }
    const float nt = fmaxf(sqrtf(red[0]), EPS_NORM);
    __syncthreads();

    float dotp = 0.f;
    for (int i = tid; i < D_DIM; i += 256) {
        float a = queries[q * D_DIM + i] / nq;   // rounded fp32 qn, stored for the GEMM
        float b = truths [q * D_DIM + i] / nt;
        qn[q * D_DIM + i] = a;
        dotp += a * b;
    }
    red[tid] = dotp; __syncthreads();
    for (int s = 128; s > 0; s >>= 1) { if (tid < s) red[tid] += red[tid + s]; __syncthreads(); }
    if (tid == 0) {
        thresh[q] = red[0];
        counts[q] = 0;
    }
}

// ---------------- kernel 2: fused fp32 WMMA GEMM + threshold count ------------------------------
__global__ __launch_bounds__(256)
void rs_gemm(const float* __restrict__ data,
             const float* __restrict__ qn,
             const float* __restrict__ thresh,
             int* __restrict__ counts,
             int N) {
    __shared__ int bc[Q_DIM];
    const int tid = threadIdx.x;
    if (tid < Q_DIM) bc[tid] = 0;
    __syncthreads();

    const int lane  = tid & 31;
    const int wave  = tid >> 5;
    const int lrow  = lane & 15;     // A: row M within wave tile; B: query column within qtile
    const int khalf = lane >> 4;     // 0 -> K={0,1}, 1 -> K={2,3} per 16x4 f32 A layout

    const long baseRow = (long)blockIdx.x * MTILE + (long)wave * 16;
    long rowA = baseRow + lrow;
    if (rowA > (long)N - 1) rowA = (long)N - 1;   // clamp tail; masked in count phase

    const float* __restrict__ aptr  = data + rowA * D_DIM + 2 * khalf;
    const float* __restrict__ bbase = qn   + (long)lrow * D_DIM + 2 * khalf;

    v8f acc[8] = {};   // 8 independent 16x16 f32 accumulator tiles (queries 0..127)

    for (int k0 = 0; k0 < D_DIM; k0 += 4) {
        const v2f a = *(const v2f*)(aptr + k0);
#pragma unroll
        for (int t = 0; t < 8; ++t) {
            const v2f b = *(const v2f*)(bbase + (long)t * 16 * D_DIM + k0);
            acc[t] = __builtin_amdgcn_wmma_f32_16x16x4_f32(
                /*neg_a=*/false, a, /*neg_b=*/false, b,
                /*c_mod=*/(short)0, acc[t], /*reuse_a=*/false, /*reuse_b=*/false);
        }
    }

    // count phase: lane holds query q = t*16 + lrow, rows M = i + 8*khalf of this wave's tile
#pragma unroll
    for (int t = 0; t < 8; ++t) {
        const int   q   = t * 16 + lrow;
        const float th  = thresh[q];
        const float tol = ATOL_C + RTOL_C * fabsf(th);
        int cnt = 0;
#pragma unroll
        for (int i = 0; i < 8; ++i) {
            const long row = baseRow + i + 8 * khalf;
            const float s  = acc[t][i];
            const bool ok  = (s >= th) || (fabsf(s - th) <= tol);
            cnt += (ok && (row < (long)N)) ? 1 : 0;
        }
        atomicAdd(&bc[q], cnt);
    }
    __syncthreads();

    if (tid < Q_DIM) {
        const int v = bc[tid];
        if (v) atomicAdd(&counts[tid], v);
    }
}

// ---------------- kernel 3: finalize (exclude self row) ----------------------------------------
__global__ void rs_final(const int* __restrict__ counts, int* __restrict__ out) {
    const int q = threadIdx.x;
    if (q < Q_DIM) out[q] = counts[q] - 1;
}

extern "C" void kernel_launch(void* const* d_in, const int* in_sizes, int n_in,
                              void* d_out, int out_size, void* d_ws, size_t ws_size,
                              hipStream_t stream) {
    const float* data    = (const float*)d_in[0];   // (N, 512) fp32
    const float* queries = (const float*)d_in[1];   // (128, 512) fp32
    const float* truths  = (const float*)d_in[2];   // (128, 512) fp32
    const int N = in_sizes[0] / D_DIM;

    // workspace layout: qn [Q*D] f32 | thresh [Q] f32 | counts [Q] i32
    float* qn     = (float*)d_ws;
    float* thresh = qn + (size_t)Q_DIM * D_DIM;
    int*   counts = (int*)(thresh + Q_DIM);

    rs_prep<<<Q_DIM, 256, 0, stream>>>(queries, truths, qn, thresh, counts);

    const int nblocks = (N + MTILE - 1) / MTILE;
    rs_gemm<<<nblocks, 256, 0, stream>>>(data, qn, thresh, counts, N);

    rs_final<<<1, Q_DIM, 0, stream>>>(counts, (int*)d_out);
}